// ScaledDotProductAttention_5308579578425
// MI455X (gfx1250) — compile-verified
//
#include <hip/hip_runtime.h>
#include <stdint.h>

// CDNA5 / gfx1250: wave32, WMMA fp32 16x16x4, fused flash-style attention
// that also materializes the normalized attention matrix.
typedef float v2f __attribute__((ext_vector_type(2)));
typedef float v8f __attribute__((ext_vector_type(8)));

namespace {
constexpr int BATCH = 32;
constexpr int SEQ   = 2048;
constexpr int DIM   = 64;
constexpr float SCALE = 0.125f;      // 1/sqrt(64)
constexpr int QT_PER_BLOCK = 8;      // 8 waves per block, one 16-row q-tile per wave
}

// Fragment layouts per CDNA5 ISA 7.12.2 (32-bit, wave32):
//   A 16x4 : lanes0-15 row=lane,   frag={K0,K1}; lanes16-31 row=lane-16, frag={K2,K3}
//   B 4x16 : lanes0-15 col=lane,   frag={K0,K1}; lanes16-31 col=lane-16, frag={K2,K3}
//   C 16x16: VGPR v -> row v (lanes0-15) / row v+8 (lanes16-31), col = lane&15
__global__ __launch_bounds__(256) void attn_fused(
    const float* __restrict__ q, const float* __restrict__ k,
    const float* __restrict__ vmat, const uint8_t* __restrict__ mask,
    float* __restrict__ attn, float* __restrict__ out)
{
  __shared__ float ptile[QT_PER_BLOCK][16][17];   // +1 pad: conflict-free transpose

  const int lane = threadIdx.x & 31;
  const int wave = threadIdx.x >> 5;
  const int half = lane >> 4;        // 0 = lanes 0-15, 1 = lanes 16-31
  const int l16  = lane & 15;
  const int b    = blockIdx.y;
  const int qr0  = (blockIdx.x * QT_PER_BLOCK + wave) * 16;
  const float NEG_INF = -__builtin_inff();

  // Q tile (16x64) as 16 A-fragments, kept in VGPRs for both passes.
  v2f aq[16];
  {
    const float* qp = q + ((size_t)b * SEQ + qr0 + l16) * DIM + 2 * half;
    #pragma unroll
    for (int i = 0; i < 16; ++i) {
      aq[i].x = qp[4 * i + 0];
      aq[i].y = qp[4 * i + 1];
    }
  }

  const uint8_t* mrow0 = mask + ((size_t)b * SEQ + qr0) * SEQ;  // 16-row mask strip

  // ---------------- Pass 1: per-LANE online softmax stats ------------------
  // Each lane tracks (m,l) over the columns it owns (1 elem/row/tile in the
  // C layout). Pure VALU/TRANS per tile: no cross-lane ops until the end.
  float m_run[8], l_run[8];
  #pragma unroll
  for (int vv = 0; vv < 8; ++vv) { m_run[vv] = NEG_INF; l_run[vv] = 0.f; }

  for (int kt = 0; kt < SEQ / 16; ++kt) {
    const float* kp = k + ((size_t)b * SEQ + kt * 16 + l16) * DIM + 2 * half;
    if (kt + 1 < SEQ / 16)
      __builtin_prefetch(kp + 16 * DIM, 0, 1);   // global_prefetch_b8, next K tile

    v8f c = {};
    #pragma unroll
    for (int i = 0; i < 16; ++i) {               // D = 64 in K-chunks of 4
      v2f bk;
      bk.x = kp[4 * i + 0];
      bk.y = kp[4 * i + 1];
      c = __builtin_amdgcn_wmma_f32_16x16x4_f32(false, aq[i], false, bk,
                                                (short)0, c, false, false);
    }

    const int col = kt * 16 + l16;
    #pragma unroll
    for (int vv = 0; vv < 8; ++vv) {
      float s = c[vv] * SCALE;
      if (mrow0[(size_t)(vv + half * 8) * SEQ + col]) s = NEG_INF;
      const float m_new = fmaxf(m_run[vv], s);
      const float a    = (s         == NEG_INF) ? 0.f : __expf(s - m_new);
      const float corr = (m_run[vv] == NEG_INF) ? 0.f : __expf(m_run[vv] - m_new);
      l_run[vv] = l_run[vv] * corr + a;
      m_run[vv] = m_new;
    }
  }

  // Merge the 16 per-lane (m,l) pairs per row: one butterfly per wave total.
  float mv[8], il[8];
  #pragma unroll
  for (int vv = 0; vv < 8; ++vv) {
    float m = m_run[vv], l = l_run[vv];
    #pragma unroll
    for (int off = 1; off < 16; off <<= 1) {     // stays inside each 16-lane half
      const float mo = __shfl_xor(m, off, 32);
      const float lo = __shfl_xor(l, off, 32);
      const float mn = fmaxf(m, mo);
      const float t0 = (m  == NEG_INF) ? 0.f : l  * __expf(m  - mn);
      const float t1 = (mo == NEG_INF) ? 0.f : lo * __expf(mo - mn);
      l = t0 + t1;
      m = mn;
    }
    mv[vv] = m;
    il[vv] = (l > 0.f) ? (1.f / l) : 0.f;
  }

  // ---------------- Pass 2: recompute, normalize, write attn, O = P V ------
  const v8f zero8 = {0.f, 0.f, 0.f, 0.f, 0.f, 0.f, 0.f, 0.f};
  v8f co[4] = {zero8, zero8, zero8, zero8};       // O tile: 16 rows x 64 cols

  for (int kt = 0; kt < SEQ / 16; ++kt) {
    const float* kp = k + ((size_t)b * SEQ + kt * 16 + l16) * DIM + 2 * half;
    const float* vbase = vmat + ((size_t)b * SEQ + kt * 16) * DIM;
    if (kt + 1 < SEQ / 16)
      __builtin_prefetch(vbase + 16 * DIM + lane * 4, 0, 1);   // next V tile

    v8f c = {};
    #pragma unroll
    for (int i = 0; i < 16; ++i) {
      v2f bk;
      bk.x = kp[4 * i + 0];
      bk.y = kp[4 * i + 1];
      c = __builtin_amdgcn_wmma_f32_16x16x4_f32(false, aq[i], false, bk,
                                                (short)0, c, false, false);
    }

    // Normalize with final stats; stream attn out (non-temporal: write-once,
    // never re-read -> keep L2 for K/V/mask), stage tile in LDS for transpose.
    const int col = kt * 16 + l16;
    #pragma unroll
    for (int vv = 0; vv < 8; ++vv) {
      const int r = vv + half * 8;
      float s = c[vv] * SCALE;
      float p;
      if (mrow0[(size_t)r * SEQ + col] || il[vv] == 0.f)
        p = 0.f;
      else
        p = __expf(s - mv[vv]) * il[vv];
      __builtin_nontemporal_store(p, &attn[((size_t)b * SEQ + qr0 + r) * SEQ + col]);
      ptile[wave][r][l16] = p;
    }
    __asm__ volatile("s_wait_dscnt 0" ::: "memory");  // DS in-order per wave; pin it

    // A fragments of P from LDS (cross-lane transpose); B fragments of V.
    #pragma unroll
    for (int j = 0; j < 4; ++j) {                 // 16 keys in K-chunks of 4
      v2f ap;
      ap.x = ptile[wave][l16][4 * j + 2 * half + 0];
      ap.y = ptile[wave][l16][4 * j + 2 * half + 1];
      const float* vp = vbase + (4 * j + 2 * half) * DIM;
      #pragma unroll
      for (int n = 0; n < 4; ++n) {               // D = 64 as 4 N-blocks of 16
        v2f bv;
        bv.x = vp[16 * n + l16];
        bv.y = vp[DIM + 16 * n + l16];
        co[n] = __builtin_amdgcn_wmma_f32_16x16x4_f32(false, ap, false, bv,
                                                      (short)0, co[n], false, false);
      }
    }
    __asm__ volatile("s_wait_dscnt 0" ::: "memory");  // reads done before next writes
  }

  // Write O (C layout -> row-major [B,S,D]).
  #pragma unroll
  for (int n = 0; n < 4; ++n) {
    #pragma unroll
    for (int vv = 0; vv < 8; ++vv) {
      const int row = qr0 + vv + half * 8;
      out[((size_t)b * SEQ + row) * DIM + 16 * n + l16] = co[n][vv];
    }
  }
}

// ---------------------------------------------------------------------------
extern "C" void kernel_launch(void* const* d_in, const int* in_sizes, int n_in,
                              void* d_out, int out_size, void* d_ws, size_t ws_size,
                              hipStream_t stream) {
  const float*   q    = (const float*)d_in[0];
  const float*   k    = (const float*)d_in[1];
  const float*   v    = (const float*)d_in[2];
  const uint8_t* mask = (const uint8_t*)d_in[3];   // jnp.bool_ -> 1 byte/elem

  float* out  = (float*)d_out;                            // [B,S,D]
  float* attn = out + (size_t)BATCH * SEQ * DIM;          // [B,S,S]

  dim3 block(32 * QT_PER_BLOCK);                          // 256 threads = 8 waves
  dim3 grid(SEQ / 16 / QT_PER_BLOCK, BATCH);              // (16, 32)

  hipLaunchKernelGGL(attn_fused, grid, block, 0, stream,
                     q, k, v, mask, attn, out);
}